// KANLayer_73306501808840
// MI455X (gfx1250) — compile-verified
//
#include <hip/hip_runtime.h>

typedef __attribute__((ext_vector_type(2))) float v2f;
typedef __attribute__((ext_vector_type(4))) float v4f;
typedef __attribute__((ext_vector_type(8))) float v8f;

#define D_IN    128
#define D_OUT   128
#define NKNOT   8
#define KTOT    (D_IN * NKNOT)       // 1024 = GEMM K dimension
#define CHUNK_K 64                   // K per LDS stage (8 inputs x 8 knots)
#define CHUNK_J (CHUNK_K / NKNOT)    // 8 inputs per chunk
#define NCHUNK  (KTOT / CHUNK_K)     // 16
#define TILE_M  64                   // rows per workgroup
#define ASTR    68                   // padded LDS stride for A (conflict-free b64)
#define BPSTR   288                  // padded LDS stride per k-PAIR row of B
                                     //   288 % 64 == 32 -> lanes16-31 hit banks 32..63

// ---------------------------------------------------------------------------
// Workspace B, pre-interleaved by k-pairs for direct b64 WMMA-fragment loads:
//   Bp[(k>>1)*256 + o*2 + (k&1)] = weights[o][j] * coeffs[o][j][kk],
// where k = j*8+kk is the GEMM K index.
// ---------------------------------------------------------------------------
__global__ __launch_bounds__(256) void kan_prep_B(
    const float* __restrict__ weights, const float* __restrict__ coeffs,
    float* __restrict__ Bp) {
  int pos = blockIdx.x * 256 + threadIdx.x;   // [0, KTOT*D_OUT)
  int kp  = pos >> 8;                         // k-pair row
  int rem = pos & 255;
  int o   = rem >> 1;
  int k   = kp * 2 + (rem & 1);
  int j   = k >> 3;
  int kk  = k & 7;
  Bp[pos] = weights[o * D_IN + j] * coeffs[(o * D_IN + j) * NKNOT + kk];
}

// ---------------------------------------------------------------------------
// out[n][o] = sum_k A[n][k] * B[k][o] + bias[o],  A built from spline basis.
// 256 threads = 8 waves; wave w: rows m0=(w>>1)*16, cols nbase=(w&1)*64.
// ---------------------------------------------------------------------------
__global__ __launch_bounds__(256) void kan_wmma(
    const float* __restrict__ x, const float* __restrict__ Bp,
    const float* __restrict__ bias, float* __restrict__ out) {
  __shared__ float sA[TILE_M * ASTR];            // 17.4 KB
  __shared__ float sB[(CHUNK_K / 2) * BPSTR];    // 36.9 KB, k-pair interleaved

  const int tid   = threadIdx.x;
  const int lane  = tid & 31;
  const int wave  = tid >> 5;
  const int row0  = blockIdx.x * TILE_M;
  const int m0    = (wave >> 1) * 16;
  const int nbase = (wave & 1) * 64;
  const int l16   = lane & 15;
  const int lh    = lane >> 4;                   // half-wave select

  v8f acc0 = {}, acc1 = {}, acc2 = {}, acc3 = {};

  // Fragment base pointers (ISA layout: k = v + 2*lh for A and B operands).
  // A element (m, k)   at sA[m*ASTR + k]                    -> v2f at k = kb+2lh
  // B element (k, n)   at sB[(k>>1)*BPSTR + n*2 + (k&1)]    -> v2f at kp = kb/2+lh
  const float* aRow  = &sA[(m0 + l16) * ASTR + 2 * lh];
  const float* bBase = &sB[lh * BPSTR + (nbase + l16) * 2];

  for (int ch = 0; ch < NCHUNK; ++ch) {
    const int jb0 = ch * CHUNK_J;

    // --- stage A chunk: scatter Catmull-Rom basis weights into dense 8-vec
    #pragma unroll
    for (int i = 0; i < 2; ++i) {
      int p  = tid + i * 256;            // 0..511 -> (row, j) pair
      int nl = p >> 3;                   // 0..63
      int jl = p & 7;                    // 0..7
      float xv = x[(row0 + nl) * D_IN + jb0 + jl];
      xv = fminf(fmaxf(xv, -1.0f), 1.0f);
      float t = (xv + 1.0f) * 3.5f;      // (x - min)/h, h = 2/(K-1) = 2/7
      int idx = (int)floorf(t);
      idx = idx < 0 ? 0 : (idx > 6 ? 6 : idx);
      float u  = t - (float)idx;
      float u2 = u * u, u3 = u2 * u;
      float b0 = 0.5f * (-u3 + 2.0f * u2 - u);
      float b1 = 0.5f * (3.0f * u3 - 5.0f * u2 + 2.0f);
      float b2 = 0.5f * (-3.0f * u3 + 4.0f * u2 + u);
      float b3 = 0.5f * (u3 - u2);
      float g[NKNOT] = {0.f, 0.f, 0.f, 0.f, 0.f, 0.f, 0.f, 0.f};
      int i0 = (idx - 1 < 0) ? 0 : idx - 1;
      int i3 = (idx + 2 > 7) ? 7 : idx + 2;
      g[i0]      += b0;                  // += handles clipped-index collisions
      g[idx]     += b1;
      g[idx + 1] += b2;
      g[i3]      += b3;
      float* dst = &sA[nl * ASTR + jl * NKNOT];
      *(v4f*)(dst)     = *(const v4f*)&g[0];
      *(v4f*)(dst + 4) = *(const v4f*)&g[4];
    }

    // --- stage B chunk (already k-pair interleaved in ws): linear float4 copy
    //     global pair-row = 256 dwords (64 float4), LDS pair-row = BPSTR dwords
    const v4f* bsrc = (const v4f*)(Bp + ch * (CHUNK_K / 2) * 2 * D_OUT);
    #pragma unroll
    for (int i = 0; i < 8; ++i) {
      int f   = tid + i * 256;           // float4 index, 0..2047
      int kpl = f >> 6;                  // 64 float4 per k-pair row
      int r4  = f & 63;
      *(v4f*)&sB[kpl * BPSTR + r4 * 4] = bsrc[f];
    }
    __syncthreads();

    // --- 16 k-steps x 4 n-subtiles of V_WMMA_F32_16X16X4_F32
    #pragma unroll
    for (int kb = 0; kb < CHUNK_K; kb += 4) {
      v2f a = *(const v2f*)&aRow[kb];                  // A[m][kb+2lh+{0,1}]
      const float* bk = &bBase[(kb >> 1) * BPSTR];     // k-pair row kb/2 (+lh)
      v2f b;
      b = *(const v2f*)&bk[0];
      acc0 = __builtin_amdgcn_wmma_f32_16x16x4_f32(false, a, false, b,
                                                   (short)0, acc0, false, false);
      b = *(const v2f*)&bk[32];                        // n0 += 16 -> +32 dwords
      acc1 = __builtin_amdgcn_wmma_f32_16x16x4_f32(false, a, false, b,
                                                   (short)0, acc1, false, false);
      b = *(const v2f*)&bk[64];
      acc2 = __builtin_amdgcn_wmma_f32_16x16x4_f32(false, a, false, b,
                                                   (short)0, acc2, false, false);
      b = *(const v2f*)&bk[96];
      acc3 = __builtin_amdgcn_wmma_f32_16x16x4_f32(false, a, false, b,
                                                   (short)0, acc3, false, false);
    }
    __syncthreads();
  }

  // --- epilogue: C/D layout m = r + 8*lh, n = lane%16; add bias, store
  auto store_tile = [&](const v8f& acc, int n0) {
    float bv = bias[n0 + l16];
    #pragma unroll
    for (int r = 0; r < 8; ++r) {
      int row = row0 + m0 + r + 8 * lh;
      out[row * D_OUT + n0 + l16] = acc[r] + bv;
    }
  };
  store_tile(acc0, nbase);
  store_tile(acc1, nbase + 16);
  store_tile(acc2, nbase + 32);
  store_tile(acc3, nbase + 48);
}

// ---------------------------------------------------------------------------
extern "C" void kernel_launch(void* const* d_in, const int* in_sizes, int n_in,
                              void* d_out, int out_size, void* d_ws, size_t ws_size,
                              hipStream_t stream) {
  const float* x       = (const float*)d_in[0];   // [N, 128]
  const float* coeffs  = (const float*)d_in[1];   // [128, 128, 8]
  const float* weights = (const float*)d_in[2];   // [128, 128]
  const float* bias    = (const float*)d_in[3];   // [128]
  float* out = (float*)d_out;                     // [N, 128]
  float* Bp  = (float*)d_ws;                      // [512 k-pairs][128 o][2] = 512 KB

  const int N = in_sizes[0] / D_IN;               // 1024

  kan_prep_B<<<(KTOT * D_OUT) / 256, 256, 0, stream>>>(weights, coeffs, Bp);
  kan_wmma<<<N / TILE_M, 256, 0, stream>>>(x, Bp, bias, out);
}